// TextLoss_14508399526477
// MI455X (gfx1250) — compile-verified
//
#include <hip/hip_runtime.h>
#include <hip/hip_bf16.h>
#include <math.h>

// ---------------- problem geometry ----------------
#define B_IMG   8
#define H_DIM   640
#define W_DIM   640
#define HW      409600          // H*W
#define I_POLY  3
#define N_POLY  64
#define P_POLY  20

#define BLOCKS_PER_IMG 100      // 100 blocks * 4096 px = 409600 px per image
#define PAD_BPI        128      // padded to a multiple of 64 for WMMA reduce
#define NBLOCKS        (B_IMG * BLOCKS_PER_IMG)   // 800
#define PAD_TOTAL      1024     // 8 * 128, padded total slots
#define PX_PER_BLOCK   4096
#define THREADS        256

// ---------------- histogram config ----------------
#define NB_CLS   2048
#define CLS_SCALE (2048.0f / 16.2f)   // BCE in [0, -log(1e-7)=16.12]
#define NB_DIS   1024                 // dis_map in [0,1)

// ---------------- workspace layout (floats) ----------------
#define OFF_CLS_CNT 0
#define OFF_CLS_SUM (NB_CLS)
#define OFF_DIS_CNT (2 * NB_CLS)
#define OFF_DIS_SUM (2 * NB_CLS + B_IMG * NB_DIS)
#define OFF_PART    (2 * NB_CLS + 2 * B_IMG * NB_DIS)
// 10 scalars per block: 0 clsPosSum,1 clsNPos,2 clsNNeg,3 normSum,4 angSum,
//                       5 angCnt,  6 disPosSum,7 disPosCnt,8 disNegSum,9 disNegCnt
#define WS_FLOATS   (OFF_PART + 10 * PAD_TOTAL)   // 30720 floats = 120 KB

typedef float v2f __attribute__((ext_vector_type(2)));
typedef float v8f __attribute__((ext_vector_type(8)));

// ---------------- kernel 0: zero the workspace (scratch is poisoned) --------
__global__ __launch_bounds__(256) void tl_init(float* __restrict__ ws) {
    int i = blockIdx.x * 256 + threadIdx.x;
    if (i < WS_FLOATS) ws[i] = 0.0f;
}

// ---------------- kernel 1: fused streaming pass ----------------
__global__ __launch_bounds__(THREADS)
void tl_main(const float* __restrict__ fy, const float* __restrict__ df,
             const float* __restrict__ dirf, const float* __restrict__ wm,
             const int* __restrict__ tmask, const int* __restrict__ trmask,
             float* __restrict__ ws) {
    __shared__ float sClsCnt[NB_CLS];
    __shared__ float sClsSum[NB_CLS];
    __shared__ float sDisCnt[NB_DIS];
    __shared__ float sDisSum[NB_DIS];
    __shared__ float sRed[10 * 8];

    const int t = threadIdx.x;
    for (int i = t; i < NB_CLS; i += THREADS) { sClsCnt[i] = 0.f; sClsSum[i] = 0.f; }
    for (int i = t; i < NB_DIS; i += THREADS) { sDisCnt[i] = 0.f; sDisSum[i] = 0.f; }
    __syncthreads();

    const int img = blockIdx.x / BLOCKS_PER_IMG;
    const int bi  = blockIdx.x % BLOCKS_PER_IMG;

    const float* p0p = fy + (size_t)(img * 4 + 0) * HW;
    const float* p1p = fy + (size_t)(img * 4 + 1) * HW;
    const float* p2p = fy + (size_t)(img * 4 + 2) * HW;
    const float* p3p = fy + (size_t)(img * 4 + 3) * HW;
    const float* dfp = df   + (size_t)img * HW;
    const float* g0p = dirf + (size_t)(img * 2 + 0) * HW;
    const float* g1p = dirf + (size_t)(img * 2 + 1) * HW;
    const float* wmp = wm   + (size_t)img * HW;
    const int*   tmp_ = tmask  + (size_t)img * HW;
    const int*   trp  = trmask + (size_t)img * HW;

    float acc[10];
#pragma unroll
    for (int s = 0; s < 10; ++s) acc[s] = 0.f;

    for (int sweep = 0; sweep < 4; ++sweep) {
        const int j = bi * PX_PER_BLOCK + sweep * 1024 + t * 4;
        const float4 v0  = *(const float4*)(p0p + j);
        const float4 v1  = *(const float4*)(p1p + j);
        const float4 v2  = *(const float4*)(p2p + j);
        const float4 v3  = *(const float4*)(p3p + j);
        const float4 vd  = *(const float4*)(dfp + j);
        const float4 vg0 = *(const float4*)(g0p + j);
        const float4 vg1 = *(const float4*)(g1p + j);
        const float4 vw  = *(const float4*)(wmp + j);
        const int4   vtm = *(const int4*)(tmp_ + j);
        const int4   vtr = *(const int4*)(trp + j);
#pragma unroll
        for (int k = 0; k < 4; ++k) {
            const float tmf = (float)(((const int*)&vtm)[k]);
            const float trf = (((const int*)&vtr)[k] > 0) ? 1.f : 0.f;

            // ---- cls BCE + OHEM histogram ----
            float p  = ((const float*)&v0)[k];
            float pc = fminf(fmaxf(p, 1e-7f), 1.f - 1e-7f);
            float l  = (trf > 0.5f) ? -logf(pc) : -log1pf(-pc);
            if (trf * tmf > 0.f) { acc[0] += l; acc[1] += 1.f; }
            if ((1.f - trf) * tmf > 0.f) {
                acc[2] += 1.f;
                int bin = min(NB_CLS - 1, (int)(l * CLS_SCALE));
                atomicAdd(&sClsCnt[bin], 1.f);
                atomicAdd(&sClsSum[bin], l);
            }

            // ---- distance map + per-image OHEM histogram ----
            float dfv = ((const float*)&vd)[k];
            float dd  = ((const float*)&v1)[k] - dfv;
            dd = dd * dd * tmf;
            if (dfv >= 0.001f) { acc[6] += dd; acc[7] += 1.f; }
            else {
                acc[8] += dd; acc[9] += 1.f;
                int bin = min(NB_DIS - 1, (int)(dd * (float)NB_DIS));
                atomicAdd(&sDisCnt[bin], 1.f);
                atomicAdd(&sDisSum[bin], dd);
            }

            // ---- flux: norm + angle ----
            float g0 = ((const float*)&vg0)[k];
            float g1 = ((const float*)&vg1)[k];
            float gl = sqrtf(g0 * g0 + g1 * g1);
            float gs = 0.999999f / (gl + 1e-9f);
            float gn0 = g0 * gs, gn1 = g1 * gs;
            float q2 = ((const float*)&v2)[k];
            float q3 = ((const float*)&v3)[k];
            float e0 = q2 - gn0, e1 = q3 - gn1;
            acc[3] += ((const float*)&vw)[k] * (e0 * e0 + e1 * e1) * tmf;
            if (tmf * trf > 0.f) {
                float pl2 = sqrtf(q2 * q2 + q3 * q3);
                float ps  = 0.999999f / (pl2 + 1e-9f);
                float pn0 = q2 * ps, pn1 = q3 * ps;
                float dot = pn0 * gn0 + pn1 * gn1;
                float npn = sqrtf(pn0 * pn0 + pn1 * pn1);
                float ngn = sqrtf(gn0 * gn0 + gn1 * gn1);
                float den = fmaxf(npn * ngn, 1e-8f);
                acc[4] += 1.f - dot / den;
                acc[5] += 1.f;
            }
        }
    }

    // ---- deterministic block reduction of the 10 scalars (wave32 shuffles) ----
    const int w = t >> 5, lane = t & 31;
#pragma unroll
    for (int s = 0; s < 10; ++s) {
        float v = acc[s];
        for (int o = 16; o > 0; o >>= 1) v += __shfl_down(v, o, 32);
        if (lane == 0) sRed[s * 8 + w] = v;
    }
    __syncthreads();
    if (t < 10) {
        float v = 0.f;
#pragma unroll
        for (int q = 0; q < 8; ++q) v += sRed[t * 8 + q];
        ws[OFF_PART + t * PAD_TOTAL + img * PAD_BPI + bi] = v;   // fixed slot, no atomics
    }

    // ---- flush LDS histograms to global (f32 atomics; counts stay exact) ----
    float* gClsCnt = ws + OFF_CLS_CNT;
    float* gClsSum = ws + OFF_CLS_SUM;
    float* gDisCnt = ws + OFF_DIS_CNT + (size_t)img * NB_DIS;
    float* gDisSum = ws + OFF_DIS_SUM + (size_t)img * NB_DIS;
    for (int i = t; i < NB_CLS; i += THREADS) {
        float c = sClsCnt[i];
        if (c != 0.f) { atomicAdd(&gClsCnt[i], c); atomicAdd(&gClsSum[i], sClsSum[i]); }
    }
    for (int i = t; i < NB_DIS; i += THREADS) {
        float c = sDisCnt[i];
        if (c != 0.f) { atomicAdd(&gDisCnt[i], c); atomicAdd(&gDisSum[i], sDisSum[i]); }
    }
}

// ---------------- WMMA f32 reduction: sum iters*64 floats at full precision --
// A = 16x4 chunk of the array, B = ones(4x16)  =>  D rows replicate row-sums;
// C chains across iterations. One v_wmma_f32_16x16x4_f32 replaces 63 adds.
__device__ __forceinline__ float wmma_sum(const float* __restrict__ src, int iters, int lane) {
    v8f c = {};
    v2f b = {1.0f, 1.0f};
    const int m  = lane & 15;
    const int ko = (lane >> 4) << 1;          // lanes 0-15: K=0,1  lanes 16-31: K=2,3
    for (int it = 0; it < iters; ++it) {
        const float* p = src + it * 64 + m * 4 + ko;
        v2f a = {p[0], p[1]};
        c = __builtin_amdgcn_wmma_f32_16x16x4_f32(false, a, false, b, (short)0, c, false, false);
    }
    float s = c[0] + c[1] + c[2] + c[3] + c[4] + c[5] + c[6] + c[7]; // rows 0-7 / 8-15
    s += __shfl_xor(s, 16, 32);                                     // all 16 rows
    return s;                                                       // every lane has total
}

__device__ __forceinline__ float hist_topk_sum(const float* __restrict__ cnt,
                                               const float* __restrict__ sum,
                                               int nb, float k) {
    float need = k, acc = 0.f;
    for (int b = nb - 1; b >= 0 && need > 0.f; --b) {
        float c = cnt[b];
        if (c <= 0.f) continue;
        if (c <= need) { acc += sum[b]; need -= c; }
        else           { acc += sum[b] * (need / c); need = 0.f; }
    }
    return acc;
}

// ---------------- kernel 2: finalize ----------------
__global__ __launch_bounds__(256)
void tl_finalize(const float* __restrict__ py_preds, const float* __restrict__ gt_points,
                 const int* __restrict__ inds, const float* __restrict__ ws,
                 float* __restrict__ out) {
    __shared__ float sG[6];
    __shared__ float sD[4][B_IMG];
    __shared__ float sPoly[256];

    const int t = threadIdx.x;
    const int lane = t & 31;

    if (t < 32) {   // wave 0, full EXEC: matrix-engine reductions of block partials
        const float* part = ws + OFF_PART;
        for (int s = 0; s < 6; ++s) {
            float v = wmma_sum(part + s * PAD_TOTAL, PAD_TOTAL / 64, lane);
            if (lane == 0) sG[s] = v;
        }
        for (int s = 0; s < 4; ++s)
            for (int im = 0; im < B_IMG; ++im) {
                float v = wmma_sum(part + (6 + s) * PAD_TOTAL + im * PAD_BPI, PAD_BPI / 64, lane);
                if (lane == 0) sD[s][im] = v;
            }
    }

    // ---- poly matching loss: 192 (i,n) pairs, min over 20 cyclic shifts ----
    float myPoly = 0.f;
    if (t < I_POLY * N_POLY) {
        const int i = t / N_POLY, n = t % N_POLY;
        const float* pp = py_preds  + (size_t)(i * N_POLY + n) * (P_POLY * 2);
        const float* gp = gt_points + (size_t)inds[n] * (P_POLY * 2);
        float best = 3.0e38f;
        for (int q = 0; q < P_POLY; ++q) {
            float a = 0.f;
            for (int p = 0; p < P_POLY; ++p) {
                int r = p + q; if (r >= P_POLY) r -= P_POLY;
                a += fabsf(pp[2 * p]     - gp[2 * r]) +
                     fabsf(pp[2 * p + 1] - gp[2 * r + 1]);
            }
            best = fminf(best, a * (1.f / (float)P_POLY));
        }
        myPoly = best * (1.f / (float)(I_POLY * N_POLY));
    }
    sPoly[t] = myPoly;
    __syncthreads();
    for (int off = 128; off > 0; off >>= 1) {
        if (t < off) sPoly[t] += sPoly[t + off];
        __syncthreads();
    }

    if (t == 0) {
        // ---- cls OHEM ----
        float nPos = sG[1], nNegTot = sG[2];
        float kneg = (nPos > 0.f) ? fminf(nNegTot, floorf(3.f * nPos)) : 100.f;
        float lossNeg = hist_topk_sum(ws + OFF_CLS_CNT, ws + OFF_CLS_SUM, NB_CLS, kneg);
        float lossPos = (nPos > 0.f) ? sG[0] : 0.f;
        float cls = (lossPos + lossNeg) / fmaxf(nPos + kneg, 1.f);

        // ---- dis loss (per-image OHEM rows) ----
        float disLoss = 0.f;
        for (int im = 0; im < B_IMG; ++im) {
            const float* dCnt = ws + OFF_DIS_CNT + (size_t)im * NB_DIS;
            const float* dSum = ws + OFF_DIS_SUM + (size_t)im * NB_DIS;
            float ps = sD[0][im], pc = sD[1][im], ns = sD[2][im], nc = sD[3][im];
            float row;
            if (pc > 0.f) {
                float posi = ps / fmaxf(pc, 1.f);
                float k3 = 3.f * pc;
                float nega;
                if (nc < k3) nega = ns / fmaxf(nc, 1.f);
                else         nega = hist_topk_sum(dCnt, dSum, NB_DIS, k3) / fmaxf(k3, 1.f);
                row = posi + nega;
            } else {
                // pos_cnt==0 => every element is "neg" => hist covers full pl
                row = hist_topk_sum(dCnt, dSum, NB_DIS, 100.f) * (1.f / 100.f);
            }
            disLoss += row;
        }
        disLoss *= 1.f / (float)B_IMG;

        float normLoss = sG[3] / (float)(B_IMG * H_DIM);   // mean over (B,H) of W-sums
        float angLoss  = sG[4] / fmaxf(sG[5], 1.f);
        float pointLoss = sPoly[0];

        out[0] = cls + 3.f * disLoss + normLoss + angLoss + 0.05f * pointLoss;
    }
}

// ---------------- host launcher ----------------
extern "C" void kernel_launch(void* const* d_in, const int* in_sizes, int n_in,
                              void* d_out, int out_size, void* d_ws, size_t ws_size,
                              hipStream_t stream) {
    (void)in_sizes; (void)n_in; (void)out_size; (void)ws_size;
    const float* fy   = (const float*)d_in[0];
    const float* py   = (const float*)d_in[1];
    const float* df   = (const float*)d_in[2];
    const float* dirf = (const float*)d_in[3];
    const float* wm   = (const float*)d_in[4];
    const float* gt   = (const float*)d_in[5];
    const int*   tm   = (const int*)d_in[6];
    const int*   tr   = (const int*)d_in[7];
    const int*   inds = (const int*)d_in[8];
    float* ws  = (float*)d_ws;
    float* out = (float*)d_out;

    hipLaunchKernelGGL(tl_init, dim3((WS_FLOATS + 255) / 256), dim3(256), 0, stream, ws);
    hipLaunchKernelGGL(tl_main, dim3(NBLOCKS), dim3(THREADS), 0, stream,
                       fy, df, dirf, wm, tm, tr, ws);
    hipLaunchKernelGGL(tl_finalize, dim3(1), dim3(256), 0, stream, py, gt, inds, ws, out);
}